// LTITRFGen_41034117545990
// MI455X (gfx1250) — compile-verified
//
#include <hip/hip_runtime.h>

// CDNA5 / gfx1250, wave32 only.
typedef __attribute__((ext_vector_type(2))) float v2f;
typedef __attribute__((ext_vector_type(8))) float v8f;

#define B_DIM   16
#define IN_DIM  16
#define N_WIN   32
#define OUT_DIM 32
#define T_DIM   4096
#define M_DIM   (OUT_DIM * N_WIN)        // 1024
#define M_TILES (M_DIM / 16)             // 64
#define T_TILES_PER_WAVE 4               // each wave covers 64 t-values
#define WAVES_PER_BLOCK  8               // 256 threads
#define T_GROUPS (T_DIM / (16 * T_TILES_PER_WAVE))  // 64

__global__ __launch_bounds__(256) void trf_wmma_kernel(
    const float* __restrict__ x,       // (B, inDim, T)
    const float* __restrict__ weight,  // (outDim, inDim, nWin)
    const float* __restrict__ bias,    // (outDim)
    float* __restrict__ out)           // (B, outDim, nWin, T)
{
  const int lane        = threadIdx.x & 31;
  const int waveInBlock = threadIdx.x >> 5;
  const int waveId      = blockIdx.x * WAVES_PER_BLOCK + waveInBlock;

  const int tGroup = waveId % T_GROUPS;
  const int mTile  = (waveId / T_GROUPS) % M_TILES;
  const int b      = waveId / (T_GROUPS * M_TILES);

  const int mBase   = mTile * 16;
  const int halfSel = lane >> 4;     // 0: lanes 0-15, 1: lanes 16-31
  const int lane15  = lane & 15;

  // ---- A operand: 16x16 weight tile as 4 chunks of 16x4 (f32 WMMA A layout).
  // Lane (0-15): row m = mBase+lane15, VGPR0 = K(4c+0), VGPR1 = K(4c+1)
  // Lane (16-31): same rows,           VGPR0 = K(4c+2), VGPR1 = K(4c+3)
  const int m = mBase + lane15;           // flattened (o,w) row
  const int o = m >> 5;                   // same for the entire 16-row tile
  const int w = m & (N_WIN - 1);
  v2f a[4];
#pragma unroll
  for (int c = 0; c < 4; ++c) {
    const int k0 = 4 * c + (halfSel ? 2 : 0);
    // weight[o][i][w] flat = o*(IN_DIM*N_WIN) + i*N_WIN + w
    a[c].x = weight[o * (IN_DIM * N_WIN) + (k0    ) * N_WIN + w];
    a[c].y = weight[o * (IN_DIM * N_WIN) + (k0 + 1) * N_WIN + w];
  }

  // Reference adds bias BEFORE the sum over inDim -> additive term 16*bias[o].
  const float biasTerm = (float)IN_DIM * bias[o];

  const float* xb      = x   + (size_t)b * IN_DIM * T_DIM;
  float*       outBase = out + (size_t)b * M_DIM * T_DIM + (size_t)mBase * T_DIM;

  const int t0base = tGroup * (16 * T_TILES_PER_WAVE);

#pragma unroll
  for (int tt = 0; tt < T_TILES_PER_WAVE; ++tt) {
    const int t0 = t0base + tt * 16;

    // ---- B operand: 4x16 tile of x, K striped across lane halves, N = lane15.
    v8f acc = {};
#pragma unroll
    for (int c = 0; c < 4; ++c) {
      const int k0 = 4 * c + (halfSel ? 2 : 0);
      v2f bv;
      bv.x = xb[(size_t)(k0    ) * T_DIM + t0 + lane15];
      bv.y = xb[(size_t)(k0 + 1) * T_DIM + t0 + lane15];
      // D = A*B + C  (V_WMMA_F32_16X16X4_F32), chained over K = 16
      acc = __builtin_amdgcn_wmma_f32_16x16x4_f32(
          /*neg_a=*/false, a[c], /*neg_b=*/false, bv,
          /*c_mod=*/(short)0, acc, /*reuse_a=*/false, /*reuse_b=*/false);
    }

    // ---- bias + non-temporal store (output is 256MB, never re-read: bypass L2 retention)
#pragma unroll
    for (int r = 0; r < 8; ++r) {
      const int mRow = r + (halfSel ? 8 : 0);   // C/D layout: VGPR r -> rows r / r+8
      float* p = outBase + (size_t)mRow * T_DIM + t0 + lane15;
      __builtin_nontemporal_store(acc[r] + biasTerm, p);
    }
  }
}

extern "C" void kernel_launch(void* const* d_in, const int* in_sizes, int n_in,
                              void* d_out, int out_size, void* d_ws, size_t ws_size,
                              hipStream_t stream) {
  const float* x      = (const float*)d_in[0];
  const float* weight = (const float*)d_in[1];
  const float* bias   = (const float*)d_in[2];
  float*       out    = (float*)d_out;

  const int totalWaves = B_DIM * M_TILES * T_GROUPS;       // 65536
  dim3 grid(totalWaves / WAVES_PER_BLOCK);                 // 8192 blocks
  dim3 block(32 * WAVES_PER_BLOCK);                        // 256 threads (8 waves)
  trf_wmma_kernel<<<grid, block, 0, stream>>>(x, weight, bias, out);
}